// one_hypergraph_40218073760223
// MI455X (gfx1250) — compile-verified
//
#include <hip/hip_runtime.h>

#define COLS 256

typedef float v2f __attribute__((ext_vector_type(2)));
typedef float v8f __attribute__((ext_vector_type(8)));

// ---------------------------------------------------------------------------
// Kernel 1: gathered row-sum of m_embeddings and pretrained_weight.
//   block = 256 threads = 4 row-groups x 64 lanes; each lane loads float4.
//   idx[] load is wave-uniform -> scalar load. Partial sums -> d_ws (per block).
// ---------------------------------------------------------------------------
__global__ __launch_bounds__(256) void k_gather_reduce(
    const int* __restrict__ idx, const float* __restrict__ emb,
    const float* __restrict__ pre, float* __restrict__ pm,
    float* __restrict__ pp, int n, int nb) {
  __shared__ float lm[4 * COLS];
  __shared__ float lp[4 * COLS];
  const int t  = threadIdx.x;
  const int g  = t >> 6;            // row group 0..3 (uniform within a wave32)
  const int c0 = (t & 63) << 2;     // column start for this lane's float4
  const int b  = blockIdx.x;

  const int per = (n + nb - 1) / nb;
  const int r0  = b * per;
  int r1 = r0 + per; if (r1 > n) r1 = n;

  float4 am = make_float4(0.f, 0.f, 0.f, 0.f);
  float4 ap = make_float4(0.f, 0.f, 0.f, 0.f);

  const int iters = (r1 > r0) ? ((r1 - r0 + 3) >> 2) : 0;
  for (int it = 0; it < iters; ++it) {
    const int r = r0 + (it << 2) + g;          // uniform per wave
    if (r < r1) {
      const size_t row = (size_t)idx[r] * COLS; // scalar load (wave-uniform)
      const float4 m = *(const float4*)(emb + row + c0);  // global_load_b128
      const float4 q = *(const float4*)(pre + row + c0);  // global_load_b128
      am.x += m.x; am.y += m.y; am.z += m.z; am.w += m.w;
      ap.x += q.x; ap.y += q.y; ap.z += q.z; ap.w += q.w;
    }
  }

  *(float4*)(lm + g * COLS + c0) = am;
  *(float4*)(lp + g * COLS + c0) = ap;
  __syncthreads();

  // combine the 4 row-groups; thread t owns column t
  const float sm = lm[t] + lm[COLS + t] + lm[2 * COLS + t] + lm[3 * COLS + t];
  const float sp = lp[t] + lp[COLS + t] + lp[2 * COLS + t] + lp[3 * COLS + t];
  pm[(size_t)b * COLS + t] = sm;
  pp[(size_t)b * COLS + t] = sp;
}

// ---------------------------------------------------------------------------
// Kernel 2: deterministic partial reduction + WMMA matvec epilogue.
//   out[j] = p[j] + (s @ W^T)[j] + N * bias[j]
//   16 waves; wave jb computes output columns j = jb*16 .. jb*16+15 with
//   V_WMMA_F32_16X16X4_F32, accumulating K=256 in 64 chunks of 4.
//   A = s broadcast down rows:  lane<16 -> {s[k0],s[k0+1]}, lane>=16 -> {s[k0+2],s[k0+3]}
//   B = W rows:                 lane<16 -> {W[j][k0],W[j][k0+1]}, lane>=16 -> +2,+3
//   D row 0 lives in C-VGPR0 lanes 0..15.
// ---------------------------------------------------------------------------
__global__ __launch_bounds__(512) void k_final(
    const float* __restrict__ pm, const float* __restrict__ pp,
    const float* __restrict__ W, const float* __restrict__ bias,
    float* __restrict__ out, int nb, float Nf) {
  __shared__ float s_s[COLS];
  __shared__ float s_p[COLS];
  const int t = threadIdx.x;

  if (t < COLS) {
    float acc = 0.f;
    for (int b = 0; b < nb; ++b) acc += pm[(size_t)b * COLS + t];
    s_s[t] = acc;
  } else {
    const int c = t - COLS;
    float acc = 0.f;
    for (int b = 0; b < nb; ++b) acc += pp[(size_t)b * COLS + c];
    s_p[c] = acc;
  }
  __syncthreads();

  const int wave = t >> 5;                 // 0..15: output column block
  const int lane = t & 31;
  const int n16  = lane & 15;
  const int kb   = (lane < 16) ? 0 : 2;    // K sub-offset per half-wave
  const int j    = wave * 16 + n16;        // output column this lane serves
  const float* Wrow = W + (size_t)j * COLS + kb;

  v8f c;
  c[0] = 0.f; c[1] = 0.f; c[2] = 0.f; c[3] = 0.f;
  c[4] = 0.f; c[5] = 0.f; c[6] = 0.f; c[7] = 0.f;

#pragma unroll 4
  for (int k0 = 0; k0 < COLS; k0 += 4) {
    v2f a, bm;
    a.x = s_s[k0 + kb];
    a.y = s_s[k0 + kb + 1];
    const float2 wv = *(const float2*)(Wrow + k0);
    bm.x = wv.x;
    bm.y = wv.y;
    c = __builtin_amdgcn_wmma_f32_16x16x4_f32(
        /*neg_a=*/false, a, /*neg_b=*/false, bm,
        /*c_mod=*/(short)0, c, /*reuse_a=*/false, /*reuse_b=*/false);
  }

  if (lane < 16) {
    out[j] = s_p[j] + c[0] + Nf * bias[j];
  }
}

// ---------------------------------------------------------------------------
// Host launcher
// ---------------------------------------------------------------------------
extern "C" void kernel_launch(void* const* d_in, const int* in_sizes, int n_in,
                              void* d_out, int out_size, void* d_ws, size_t ws_size,
                              hipStream_t stream) {
  const int*   idx  = (const int*)d_in[0];     // medicine_it [N]
  const float* emb  = (const float*)d_in[1];   // m_embeddings [VOCAB,256]
  const float* pre  = (const float*)d_in[2];   // pretrained_weight [VOCAB,256]
  const float* W    = (const float*)d_in[3];   // W [256,256]
  const float* bias = (const float*)d_in[4];   // bias [256]
  float* out = (float*)d_out;                  // [256]

  const int n = in_sizes[0];

  int nb = 512;                                // partial-sum blocks (1 MB ws)
  const size_t per_blk = 2 * COLS * sizeof(float);
  if ((size_t)nb * per_blk > ws_size) nb = (int)(ws_size / per_blk);
  if (nb < 1) nb = 1;

  float* pm = (float*)d_ws;                    // [nb][256]
  float* pp = pm + (size_t)nb * COLS;          // [nb][256]

  k_gather_reduce<<<nb, 256, 0, stream>>>(idx, emb, pre, pm, pp, n, nb);
  k_final<<<1, 512, 0, stream>>>(pm, pp, W, bias, out, nb, (float)n);
}